// GCN2_1348619731440
// MI455X (gfx1250) — compile-verified
//
#include <hip/hip_runtime.h>

typedef __attribute__((ext_vector_type(2))) float v2f;
typedef __attribute__((ext_vector_type(8))) float v8f;

// ---------------- degree / normalization ----------------
__global__ void k_deg_init(float* __restrict__ deg, int N) {
  int i = blockIdx.x * blockDim.x + threadIdx.x;
  if (i < N) deg[i] = 1.0f;  // self-loop contributes 1 to in-degree
}

__global__ void k_deg_count(const long long* __restrict__ dst, float* __restrict__ deg,
                            long long E) {
  long long e = (long long)blockIdx.x * blockDim.x + threadIdx.x;
  if (e < E) atomicAdd(&deg[dst[e]], 1.0f);
}

__global__ void k_deg_rsqrt(float* __restrict__ deg, int N) {
  int i = blockIdx.x * blockDim.x + threadIdx.x;
  if (i < N) {
    float d = deg[i];
    deg[i] = (d > 0.0f) ? rsqrtf(d) : 0.0f;  // becomes dinv in place
  }
}

// ---------------- dense GEMM via V_WMMA_F32_16X16X4_F32 ----------------
// One wave32 computes a full 16 x NOUT stripe of O = A(MxK) * W(KxNOUT).
// W is staged once per block into LDS, K-pair interleaved and n-padded to a
// multiple of 16 (zeros) -> each B fragment is one 64-bit LDS read and the
// inner loop has no divergent selects. Pair-rows padded +16B to avoid
// lanes0-15 / lanes16-31 bank aliasing. Per k-step: all NT B fragments are
// loaded first (one ds clause, one dscnt wait), then NT WMMAs issue
// back-to-back on independent accumulators reusing the single A fragment.
// A frag: lane L holds A[m][2*(L>>4)], A[m][2*(L>>4)+1], m = L&15 (one b64).
// B frag mirrored: lane L holds W[2*(L>>4)][n], W[2*(L>>4)+1][n], n = L&15.
// D: VGPR r -> row 16*tm + r + 8*(L>>4), col = n.
template<int K, int NOUT>
__global__ void k_gemm_wmma(const float* __restrict__ A, const float* __restrict__ W,
                            float* __restrict__ O, int M, int Mtiles) {
  constexpr int NT    = (NOUT + 15) / 16;
  constexpr int NPAD  = NT * 16;         // padded column count
  constexpr int ROWF2 = NPAD + 2;        // pair-row stride in float2 units
  __shared__ float LW[(K / 2) * ROWF2 * 2];

  // ---- stage W into LDS: LW2[p][n] = (W[2p][n], W[2p+1][n]), zero-padded n ----
  for (int idx = threadIdx.x; idx < K * NPAD; idx += blockDim.x) {
    const int k = idx / NPAD;
    const int n = idx - k * NPAD;
    const float v = ((NOUT & 15) == 0 || n < NOUT) ? W[(long long)k * NOUT + n] : 0.0f;
    LW[((k >> 1) * ROWF2 + n) * 2 + (k & 1)] = v;
  }
  __syncthreads();

  const int wave = threadIdx.x >> 5;
  const int lane = threadIdx.x & 31;
  const int tm = blockIdx.x * (blockDim.x >> 5) + wave;
  if (tm >= Mtiles) return;                 // wave-uniform: EXEC stays all-ones

  const int nlo = lane & 15;
  const int kb  = (lane >> 4) << 1;         // 0 or 2
  int mrow = tm * 16 + nlo;
  if (mrow >= M) mrow = M - 1;              // only matters for a ragged tail tile
  const float2* __restrict__ arow = (const float2*)(A + (long long)mrow * K);
  const float2* __restrict__ lw2  = (const float2*)LW;

  v8f acc[NT];
#pragma unroll
  for (int t = 0; t < NT; ++t) acc[t] = (v8f){};

#pragma unroll 4
  for (int k0 = 0; k0 < K; k0 += 4) {
    const int ka = k0 + kb;
    const float2 af = arow[ka >> 1];        // global_load_b64
    v2f a; a.x = af.x; a.y = af.y;
    const float2* __restrict__ brow = lw2 + (ka >> 1) * ROWF2 + nlo;

    // phase 1: fetch all B fragments (one LDS clause, one wait)
    float2 bf[NT];
#pragma unroll
    for (int t = 0; t < NT; ++t) bf[t] = brow[t * 16];   // ds_load_(2addr_)b64

    // phase 2: back-to-back WMMAs, A fragment reused, no RAW hazards
#pragma unroll
    for (int t = 0; t < NT; ++t) {
      v2f b; b.x = bf[t].x; b.y = bf[t].y;
      // (neg_a, A, neg_b, B, c_mod, C, reuse_a, reuse_b)
      acc[t] = __builtin_amdgcn_wmma_f32_16x16x4_f32(false, a, false, b, (short)0,
                                                     acc[t], false, false);
    }
  }

  const int rbase = tm * 16 + ((lane >> 4) << 3);
  const bool full_m = (tm * 16 + 16) <= M;  // wave-uniform fast path
  if (full_m) {
#pragma unroll
    for (int t = 0; t < NT; ++t) {
      const int ncol = t * 16 + nlo;
      if ((NOUT & 15) == 0 || ncol < NOUT) {  // one branch per n-tile, not per row
        float* __restrict__ ocol = O + (long long)rbase * NOUT + ncol;
#pragma unroll
        for (int r = 0; r < 8; ++r) ocol[(long long)r * NOUT] = acc[t][r];
      }
    }
  } else {
#pragma unroll
    for (int t = 0; t < NT; ++t) {
      const int ncol = t * 16 + nlo;
#pragma unroll
      for (int r = 0; r < 8; ++r)
        if (rbase + r < M && ((NOUT & 15) == 0 || ncol < NOUT))
          O[(long long)(rbase + r) * NOUT + ncol] = acc[t][r];
    }
  }
}

// ---------------- aggregation ----------------
// out[i,f] = dinv[i]^2 * h[i,f]  (self-loop term; also initializes out)
template<int F>
__global__ void k_selfloop(const float* __restrict__ h, const float* __restrict__ dinv,
                           float* __restrict__ out, long long total) {
  long long i = (long long)blockIdx.x * blockDim.x + threadIdx.x;
  if (i >= total) return;
  const int node = (int)(i / F);
  const float d = dinv[node];
  out[i] = d * d * h[i];
}

// F=128: one wave32 per edge; lane handles 4 consecutive floats (float4 gather,
// 4x global_atomic_add_f32 scatter — all L2 resident).
__global__ void k_scatter128(const long long* __restrict__ src,
                             const long long* __restrict__ dst,
                             const float* __restrict__ dinv,
                             const float* __restrict__ h,
                             float* __restrict__ out, long long E) {
  const long long gid = (long long)blockIdx.x * blockDim.x + threadIdx.x;
  const long long e = gid >> 5;
  const int lane = threadIdx.x & 31;
  if (e >= E) return;
  const long long s = src[e];
  const long long d = dst[e];
  const float nrm = dinv[s] * dinv[d];
  const float4 v = ((const float4*)(h + s * 128))[lane];
  float* orow = out + d * 128 + lane * 4;
  atomicAdd(orow + 0, nrm * v.x);
  atomicAdd(orow + 1, nrm * v.y);
  atomicAdd(orow + 2, nrm * v.z);
  atomicAdd(orow + 3, nrm * v.w);
}

// F=40: 8 threads per edge, 5 floats each.
__global__ void k_scatter40(const long long* __restrict__ src,
                            const long long* __restrict__ dst,
                            const float* __restrict__ dinv,
                            const float* __restrict__ h,
                            float* __restrict__ out, long long E) {
  const long long gid = (long long)blockIdx.x * blockDim.x + threadIdx.x;
  const long long e = gid >> 3;
  const int j = (int)(gid & 7);
  if (e >= E) return;
  const long long s = src[e];
  const long long d = dst[e];
  const float nrm = dinv[s] * dinv[d];
  const float* hrow = h + s * 40 + j * 5;
  float* orow = out + d * 40 + j * 5;
#pragma unroll
  for (int t = 0; t < 5; ++t) atomicAdd(orow + t, nrm * hrow[t]);
}

template<int F, bool RELU>
__global__ void k_bias(float* __restrict__ h, const float* __restrict__ b, long long total) {
  long long i = (long long)blockIdx.x * blockDim.x + threadIdx.x;
  if (i >= total) return;
  const int f = (int)(i % F);
  float v = h[i] + b[f];
  h[i] = RELU ? fmaxf(v, 0.0f) : v;
}

// ---------------- launch ----------------
extern "C" void kernel_launch(void* const* d_in, const int* in_sizes, int n_in,
                              void* d_out, int out_size, void* d_ws, size_t ws_size,
                              hipStream_t stream) {
  const float*     x   = (const float*)d_in[0];
  const long long* ei  = (const long long*)d_in[1];   // int64 [2,E] flat
  const float*     W1  = (const float*)d_in[2];
  const float*     b1  = (const float*)d_in[3];
  const float*     W2  = (const float*)d_in[4];
  const float*     b2  = (const float*)d_in[5];
  float*           out = (float*)d_out;

  const int  FIN = 64, H = 128, C = 40;
  const int  N   = in_sizes[0] / FIN;
  const long long E = (long long)in_sizes[1] / 2;
  const long long* src = ei;
  const long long* dst = ei + E;

  // workspace carve-out (floats): dinv | B1 (N*H) | B2 (N*H) | B3 (N*C)
  float* ws   = (float*)d_ws;
  float* dinv = ws;
  long long off = ((long long)N + 63) & ~63LL;
  float* B1 = ws + off;
  float* B2 = B1 + (long long)N * H;
  float* B3 = B2 + (long long)N * H;

  const dim3 blk(256);
  const long long NH = (long long)N * H;
  const long long NC = (long long)N * C;

  // symmetric normalization
  k_deg_init <<<(N + 255) / 256, blk, 0, stream>>>(dinv, N);
  k_deg_count<<<(unsigned)((E + 255) / 256), blk, 0, stream>>>(dst, dinv, E);
  k_deg_rsqrt<<<(N + 255) / 256, blk, 0, stream>>>(dinv, N);

  const int Mtiles = (N + 15) / 16;       // one wave per 16-row stripe

  // ---- layer 1: h1 = relu( A_hat (x W1) + b1 ) ----
  k_gemm_wmma<64, 128><<<(Mtiles + 7) / 8, blk, 0, stream>>>(x, W1, B1, N, Mtiles);
  k_selfloop<128><<<(unsigned)((NH + 255) / 256), blk, 0, stream>>>(B1, dinv, B2, NH);
  k_scatter128<<<(unsigned)((E * 32 + 255) / 256), blk, 0, stream>>>(src, dst, dinv, B1, B2, E);
  k_bias<128, true><<<(unsigned)((NH + 255) / 256), blk, 0, stream>>>(B2, b1, NH);

  // ---- layer 2: out = A_hat (h1 W2) + b2 ----
  k_gemm_wmma<128, 40><<<(Mtiles + 7) / 8, blk, 0, stream>>>(B2, W2, B3, N, Mtiles);
  k_selfloop<40><<<(unsigned)((NC + 255) / 256), blk, 0, stream>>>(B3, dinv, out, NC);
  k_scatter40<<<(unsigned)((E * 8 + 255) / 256), blk, 0, stream>>>(src, dst, dinv, B3, out, E);
  k_bias<40, false><<<(unsigned)((NC + 255) / 256), blk, 0, stream>>>(out, b2, NC);
}